// P2RNVisibilityRebuildModule_57251914056192
// MI455X (gfx1250) — compile-verified
//
#include <hip/hip_runtime.h>
#include <math.h>

#define PLANE 65536   // 256*256
#define NKPT  68

typedef __attribute__((ext_vector_type(2))) float v2f;
typedef __attribute__((ext_vector_type(8))) float v8f;

// ---------------------------------------------------------------------------
// Stage A: per-batch keypoint gather + weighted Kabsch-style solve (no
// reflection fix, matching the reference R = V @ U^T exactly).
// Writes 12 floats per batch to RT: [0..8] = s*R row-major, [9..11] = t.
// ---------------------------------------------------------------------------
__global__ __launch_bounds__(128) void solve_kernel(
    const float* __restrict__ Offset,
    const float* __restrict__ Posmap,
    const float* __restrict__ att,
    const float* __restrict__ meanp,
    const int*   __restrict__ uv,
    float* __restrict__ RT)
{
    __shared__ float s_src[NKPT][3];
    __shared__ float s_dst[NKPT][3];
    __shared__ float s_w[NKPT];

    const int b = blockIdx.x;
    const int k = threadIdx.x;
    const size_t pb = (size_t)b * 3 * PLANE;

    if (k < NKPT) {
        const int u = uv[2 * k + 0];   // indexes W dim
        const int v = uv[2 * k + 1];   // indexes H dim
        const int pix = u * 256 + v;

        const float p0 = Posmap[pb + 0 * PLANE + pix];
        const float p1 = Posmap[pb + 1 * PLANE + pix];
        const float p2 = Posmap[pb + 2 * PLANE + pix];
        s_dst[k][0] = p0; s_dst[k][1] = p1; s_dst[k][2] = p2;

        s_src[k][0] = Offset[pb + 0 * PLANE + pix] * 6.0f + meanp[0 * PLANE + pix];
        s_src[k][1] = Offset[pb + 1 * PLANE + pix] * 6.0f + meanp[1 * PLANE + pix];
        s_src[k][2] = Offset[pb + 2 * PLANE + pix] * 6.0f + meanp[2 * PLANE + pix];

        // visibility: z = (p - p_left) x (p_up - p_left) on channels 0,1
        float pl0 = 0.f, pl1 = 0.f, pu0 = 0.f, pu1 = 0.f;
        if (u > 0) {
            const int q = (u - 1) * 256 + v;
            pl0 = Posmap[pb + 0 * PLANE + q];
            pl1 = Posmap[pb + 1 * PLANE + q];
        }
        if (v > 0) {
            const int q = u * 256 + (v - 1);
            pu0 = Posmap[pb + 0 * PLANE + q];
            pu1 = Posmap[pb + 1 * PLANE + q];
        }
        const float ab0 = p0 - pl0, ab1 = p1 - pl1;
        const float bc0 = pu0 - pl0, bc1 = pu1 - pl1;
        const float z = ab0 * bc1 - ab1 * bc0;
        float w = (z > 0.f) ? 1.0f : ((z < 0.f) ? 0.1f : z);

        // attention weight: indices from dst channels (x -> ix, y -> iy)
        int ix = (int)(p0 * 280.0f);   // trunc toward zero; values >= 0
        int iy = (int)(p1 * 280.0f);
        ix >>= 3; iy >>= 3;            // floor div 8 (non-negative)
        w *= att[(size_t)b * 35 * 35 + iy * 35 + ix];
        s_w[k] = w;
    }
    __syncthreads();

    if (threadIdx.x != 0) return;

    // d1/d2: sum of distances to keypoint 33
    float d1 = 0.f, d2 = 0.f;
    {
        const float a0 = s_src[33][0], a1 = s_src[33][1], a2 = s_src[33][2];
        const float c0 = s_dst[33][0], c1 = s_dst[33][1], c2 = s_dst[33][2];
        for (int i = 0; i < NKPT; ++i) {
            float e0 = s_src[i][0] - a0, e1 = s_src[i][1] - a1, e2 = s_src[i][2] - a2;
            d1 += sqrtf(e0 * e0 + e1 * e1 + e2 * e2);
            float f0 = s_dst[i][0] - c0, f1 = s_dst[i][1] - c1, f2 = s_dst[i][2] - c2;
            d2 += sqrtf(f0 * f0 + f1 * f1 + f2 * f2);
        }
    }
    const float s = d2 / d1;

    // means of A = src*s and B = dst
    float Am[3] = {0.f, 0.f, 0.f};
    float Bm[3] = {0.f, 0.f, 0.f};
    for (int i = 0; i < NKPT; ++i)
        for (int c = 0; c < 3; ++c) {
            Am[c] += s_src[i][c];
            Bm[c] += s_dst[i][c];
        }
    for (int c = 0; c < 3; ++c) {
        Am[c] = Am[c] * s * (1.0f / NKPT);
        Bm[c] = Bm[c] * (1.0f / NKPT);
    }

    // H[i][j] = sum_k w_k * (src_k[i]*s - Am[i]) * (dst_k[j] - Bm[j])
    float H[3][3] = {{0.f}};
    for (int kk = 0; kk < NKPT; ++kk) {
        const float wk = s_w[kk];
        float aa[3], bb[3];
        for (int c = 0; c < 3; ++c) {
            aa[c] = s_src[kk][c] * s - Am[c];
            bb[c] = s_dst[kk][c] - Bm[c];
        }
        for (int i = 0; i < 3; ++i)
            for (int j = 0; j < 3; ++j)
                H[i][j] += wk * aa[i] * bb[j];
    }

    // G = H^T H (symmetric), Jacobi eigen-decomposition -> Vm (columns = v_k)
    float G[3][3];
    for (int i = 0; i < 3; ++i)
        for (int j = 0; j < 3; ++j)
            G[i][j] = H[0][i] * H[0][j] + H[1][i] * H[1][j] + H[2][i] * H[2][j];

    float Vm[3][3] = {{1.f, 0.f, 0.f}, {0.f, 1.f, 0.f}, {0.f, 0.f, 1.f}};
    const int JP[3] = {0, 0, 1};
    const int JQ[3] = {1, 2, 2};
    for (int sweep = 0; sweep < 12; ++sweep) {
        for (int r = 0; r < 3; ++r) {
            const int p = JP[r], q = JQ[r];
            const float apq = G[p][q];
            if (fabsf(apq) > 1e-20f) {
                const float theta = (G[q][q] - G[p][p]) / (2.0f * apq);
                float tq = 1.0f / (fabsf(theta) + sqrtf(1.0f + theta * theta));
                if (theta < 0.0f) tq = -tq;
                const float cq = 1.0f / sqrtf(1.0f + tq * tq);
                const float sq = tq * cq;
                for (int m = 0; m < 3; ++m) {           // G <- G J
                    const float gp = G[m][p], gq = G[m][q];
                    G[m][p] = cq * gp - sq * gq;
                    G[m][q] = sq * gp + cq * gq;
                }
                for (int m = 0; m < 3; ++m) {           // G <- J^T G
                    const float gp = G[p][m], gq = G[q][m];
                    G[p][m] = cq * gp - sq * gq;
                    G[q][m] = sq * gp + cq * gq;
                }
                for (int m = 0; m < 3; ++m) {           // V <- V J
                    const float vp = Vm[m][p], vq = Vm[m][q];
                    Vm[m][p] = cq * vp - sq * vq;
                    Vm[m][q] = sq * vp + cq * vq;
                }
            }
        }
    }

    // U columns: u_k = H v_k / ||H v_k||  (||H v_k|| = sigma_k)
    float U[3][3];   // U[j][k] = component j of u_k
    for (int kc = 0; kc < 3; ++kc) {
        float u0 = H[0][0] * Vm[0][kc] + H[0][1] * Vm[1][kc] + H[0][2] * Vm[2][kc];
        float u1 = H[1][0] * Vm[0][kc] + H[1][1] * Vm[1][kc] + H[1][2] * Vm[2][kc];
        float u2 = H[2][0] * Vm[0][kc] + H[2][1] * Vm[1][kc] + H[2][2] * Vm[2][kc];
        float n = sqrtf(u0 * u0 + u1 * u1 + u2 * u2);
        n = fmaxf(n, 1e-30f);
        U[0][kc] = u0 / n; U[1][kc] = u1 / n; U[2][kc] = u2 / n;
    }

    // R = V U^T ; t = Bmean - R * Amean
    float R[3][3];
    for (int i = 0; i < 3; ++i)
        for (int j = 0; j < 3; ++j)
            R[i][j] = Vm[i][0] * U[j][0] + Vm[i][1] * U[j][1] + Vm[i][2] * U[j][2];

    float* o = RT + b * 12;
    for (int i = 0; i < 3; ++i) {
        for (int j = 0; j < 3; ++j)
            o[i * 3 + j] = s * R[i][j];           // scaled rotation
        o[9 + i] = Bm[i] - (R[i][0] * Am[0] + R[i][1] * Am[1] + R[i][2] * Am[2]);
    }
}

// ---------------------------------------------------------------------------
// Stage B: out[b,i,p] = sum_j (Offset[b,j,p]*6 + mean[j,p]) * sR[i][j] + t[i]
// via V_WMMA_F32_16X16X4_F32, branch-free inner loop:
//   A (16x4): rows 0..2 = sR, rest 0.  Lane m<16: (R[m][0],R[m][1]);
//             lane>=16: (R[m][2],0).   (ISA 32-bit A 16x4: VGPR0=K0/K2, V1=K1/K3)
//   B (4x16): column = point. bx loaded with plane baked into per-lane base
//             address (lo->plane0, hi->plane2); by loaded as one contiguous
//             128B wave load covering both 16-point groups, split via shfl.
//   C: rows 0..2 = t broadcast.
//   Two WMMAs per iteration (32 points); D tiles merged with shfl_up+cndmask
//   so all three channel stores are full-wave contiguous 128B stores.
// ---------------------------------------------------------------------------
__global__ __launch_bounds__(256) void apply_kernel(
    const float* __restrict__ Offset,
    const float* __restrict__ meanp,
    const float* __restrict__ RT,
    float* __restrict__ out)
{
    const int blk   = blockIdx.x;
    const int b     = blk >> 4;                 // 16 blocks per batch
    const int chunk = (blk & 15) * 4096;        // 4096 points per block
    const int lane  = threadIdx.x & 31;
    const int wave  = threadIdx.x >> 5;
    const int m     = lane & 15;
    const bool hi   = lane >= 16;

    const float* rt = RT + b * 12;

    // A-matrix per-lane values (rotation rows, zero-padded)
    v2f a;
    a[0] = 0.0f; a[1] = 0.0f;
    if (m < 3) {
        a[0] = hi ? rt[m * 3 + 2] : rt[m * 3 + 0];
        a[1] = hi ? 0.0f          : rt[m * 3 + 1];
    }

    // C-matrix: translation broadcast on rows 0..2 (lanes<16 -> M = VGPR idx)
    v8f cacc = {};
    if (!hi) {
        cacc[0] = rt[9];
        cacc[1] = rt[10];
        cacc[2] = rt[11];
    }

    const float* off_b = Offset + (size_t)b * 3 * PLANE;
    float*       out_b = out    + (size_t)b * 3 * PLANE;

    const int wbase = chunk + wave * 512;       // 512 points per wave
    const int cA    = hi ? 2 : 0;               // bx plane baked into address

    const float* pX  = off_b + cA * PLANE + wbase + m;     // bx offsets
    const float* pMX = meanp + cA * PLANE + wbase + m;     // bx means
    const float* pY  = off_b + 1 * PLANE + wbase + lane;   // by, contiguous 128B
    const float* pMY = meanp + 1 * PLANE + wbase + lane;
    float*       pO  = out_b + wbase + lane;               // stores, contiguous

#pragma unroll 4
    for (int it = 0; it < 16; ++it) {
        const int o = it * 32;                  // 32 points per iteration

        // group A (points o..o+15) and group B (o+16..o+31) bx values
        const float bxA = fmaf(pX[o],      6.0f, pMX[o]);
        const float bxB = fmaf(pX[o + 16], 6.0f, pMX[o + 16]);
        // channel-1 for both groups in one contiguous wave load
        const float byF = fmaf(pY[o], 6.0f, pMY[o]);
        const float byBr = __shfl_down(byF, 16, 32);
        const float byA = hi ? 0.0f : byF;      // hi lanes carry K=3 pad (zero)
        const float byB = hi ? 0.0f : byBr;

        v2f bvA; bvA[0] = bxA; bvA[1] = byA;
        v2f bvB; bvB[0] = bxB; bvB[1] = byB;

        const v8f dA = __builtin_amdgcn_wmma_f32_16x16x4_f32(
            false, a, false, bvA, (short)0, cacc, false, false);
        const v8f dB = __builtin_amdgcn_wmma_f32_16x16x4_f32(
            false, a, false, bvB, (short)0, cacc, false, false);

        // merge: lanes 0..15 take dA row i, lanes 16..31 take dB row i
#pragma unroll
        for (int i = 0; i < 3; ++i) {
            const float hiv  = __shfl_up(dB[i], 16, 32);
            const float comb = hi ? hiv : dA[i];
            pO[i * PLANE + o] = comb;           // full-wave 128B store
        }
    }
}

// ---------------------------------------------------------------------------
extern "C" void kernel_launch(void* const* d_in, const int* in_sizes, int n_in,
                              void* d_out, int out_size, void* d_ws, size_t ws_size,
                              hipStream_t stream) {
    const float* Offset = (const float*)d_in[0];   // (B,3,256,256) f32
    const float* Posmap = (const float*)d_in[1];   // (B,3,256,256) f32
    const float* att    = (const float*)d_in[2];   // (B,1,35,35)   f32
    const float* meanp  = (const float*)d_in[3];   // (3,256,256)   f32
    const int*   uv     = (const int*)d_in[4];     // (68,2)        i32
    float*       out    = (float*)d_out;           // (B,3,256,256) f32
    float*       RT     = (float*)d_ws;            // B * 12 floats scratch

    const int B = in_sizes[0] / (3 * PLANE);       // 64

    solve_kernel<<<B, 128, 0, stream>>>(Offset, Posmap, att, meanp, uv, RT);
    apply_kernel<<<B * 16, 256, 0, stream>>>(Offset, meanp, RT, out);
}